// MyLoss_33045478375584
// MI455X (gfx1250) — compile-verified
//
#include <hip/hip_runtime.h>
#include <stdint.h>

#define NCH         30
#define TCELLS      128
#define TILE_ELEMS  (TCELLS * NCH)           // 3840 floats = 15360 bytes
#define NCELL       (8192 * 7 * 7)           // 401408
#define NTILES      (NCELL / TCELLS)         // 3136 (exact)
#define TILES_PER_BLOCK 4
#define NBLOCKS     (NTILES / TILES_PER_BLOCK) // 784 (exact)
#define INV_BATCH   (1.0f / 8192.0f)

typedef unsigned int u32x4 __attribute__((ext_vector_type(4)));
typedef int          i32x8 __attribute__((ext_vector_type(8)));
typedef int          i32x4 __attribute__((ext_vector_type(4)));

// Issue one TDM descriptor: contiguous 1-D tile of TILE_ELEMS f32 from global -> LDS.
// D# layout per CDNA5 ISA (verified in asm round 2).
__device__ __forceinline__ void tdm_load_tile(const float* gptr, uint32_t lds_off) {
    uint64_t ga = (uint64_t)(uintptr_t)gptr;
    u32x4 g0;
    g0.x = 1u;                                              // count=1, user descriptor
    g0.y = lds_off;                                         // lds_addr (bytes)
    g0.z = (uint32_t)ga;                                    // global_addr[31:0]
    g0.w = (uint32_t)((ga >> 32) & 0x01FFFFFFu) | (2u << 30); // global_addr[56:32], type=2
    i32x8 g1;
    g1.s0 = (int)(2u << 16);                                // data_size = 2 (4 bytes)
    g1.s1 = (int)(((uint32_t)TILE_ELEMS & 0xFFFFu) << 16);  // tensor_dim0[15:0] @bits63:48
    g1.s2 = (int)((((uint32_t)TILE_ELEMS >> 16) & 0xFFFFu)  // tensor_dim0[31:16]
                  | (1u << 16));                            // tensor_dim1[15:0] = 1
    g1.s3 = (int)(((uint32_t)TILE_ELEMS & 0xFFFFu) << 16);  // tile_dim0 @bits127:112
    g1.s4 = 1;                                              // tile_dim1=1, tile_dim2=0
    g1.s5 = (int)TILE_ELEMS;                                // tensor_dim0_stride[31:0]
    g1.s6 = 0;
    g1.s7 = 0;
    i32x4 z4 = {0, 0, 0, 0};                                // groups 2/3 unused (<=2D tile)
    i32x8 z8 = {0, 0, 0, 0, 0, 0, 0, 0};
    __builtin_amdgcn_tensor_load_to_lds(g0, g1, z4, z4, z8, 0);
}

// Per-cell YOLO loss, mirrors the JAX reference float-for-float.
__device__ __forceinline__ float cell_loss(const float* __restrict__ P,
                                           const float* __restrict__ G) {
    const float gx = G[0], gy = G[1], gw = G[2], gh = G[3], gobj = G[4];
    const float objf   = (gobj > 0.0f) ? 1.0f : 0.0f;
    const float noobjf = 1.0f - objf;

    // ground-truth box corners (half = 0.5*GRID*wh = 3.5*wh)
    const float ghw = 3.5f * gw, ghh = 3.5f * gh;
    const float gx1 = gx - ghw, gy1 = gy - ghh;
    const float gx2 = gx + ghw, gy2 = gy + ghh;
    const float ga  = (gx2 - gx1) * (gy2 - gy1);

    float iou[2], conf[2], bxv[2], byv[2], bwv[2], bhv[2];
#pragma unroll
    for (int b = 0; b < 2; ++b) {
        const int o = 5 * b;
        const float bx = P[o + 0], by = P[o + 1], bw = P[o + 2], bh = P[o + 3];
        conf[b] = P[o + 4];
        bxv[b] = bx; byv[b] = by; bwv[b] = bw; bhv[b] = bh;
        const float hw = 3.5f * bw, hh = 3.5f * bh;
        const float x1 = bx - hw, y1 = by - hh, x2 = bx + hw, y2 = by + hh;
        float iw = fminf(x2, gx2) - fmaxf(x1, gx1);
        float ih = fminf(y2, gy2) - fmaxf(y1, gy1);
        iw = fmaxf(iw, 0.0f); ih = fmaxf(ih, 0.0f);
        const float inter = iw * ih;
        const float a1 = (x2 - x1) * (y2 - y1);
        iou[b] = inter / (a1 + ga - inter);
    }

    // jnp.argmax returns first max -> box 1 responsible only on strict greater
    const int   r       = (iou[1] > iou[0]) ? 1 : 0;
    const float max_iou = fmaxf(iou[0], iou[1]);

    const float dx = bxv[r] - gx, dy = byv[r] - gy;
    const float dw = sqrtf(bwv[r]) - sqrtf(gw);
    const float dh = sqrtf(bhv[r]) - sqrtf(gh);
    const float coord = dx * dx + dy * dy + dw * dw + dh * dh;

    const float dresp       = conf[r] - max_iou;
    const float resp_loss   = dresp * dresp;
    const float cirr        = conf[1 - r];
    const float irresp_loss = cirr * cirr;

    const float d4 = P[4] - gobj, d9 = P[9] - gobj;
    const float noobj_term = d4 * d4 + d9 * d9;

    float cls = 0.0f;
#pragma unroll
    for (int c = 10; c < 30; ++c) {
        const float d = P[c] - G[c];
        cls += d * d;
    }

    // total = 5*coord + response + 0.5*noobj + class + response + irresponse
    return objf * (5.0f * coord + 2.0f * resp_loss + irresp_loss + cls)
         + 0.5f * noobjf * noobj_term;
}

__global__ void yolo_loss_partial(const float* __restrict__ pred,
                                  const float* __restrict__ gt,
                                  float* __restrict__ ws) {
    __shared__ float s_pred[2][TILE_ELEMS];
    __shared__ float s_gt[2][TILE_ELEMS];
    __shared__ float s_part[TCELLS / 32];

    const int  tile0  = blockIdx.x * TILES_PER_BLOCK;
    const bool issuer = (threadIdx.x < 32u);

    // Prime the pipeline: tile 0 -> buffer 0.
    if (issuer) {
        const size_t b0 = (size_t)tile0 * TILE_ELEMS;
        tdm_load_tile(pred + b0, (uint32_t)(uintptr_t)&s_pred[0][0]);
        tdm_load_tile(gt   + b0, (uint32_t)(uintptr_t)&s_gt[0][0]);
    }

    float acc = 0.0f;

#pragma unroll
    for (int t = 0; t < TILES_PER_BLOCK; ++t) {
        const int buf = t & 1;
        if (issuer) {
            if (t + 1 < TILES_PER_BLOCK) {
                // Prefetch tile t+1 into the other buffer (safe: last read of that
                // buffer finished before the end-of-iteration barrier of t-1),
                // then wait only for tile t's two loads (TENSORcnt is in-order).
                const size_t bn = (size_t)(tile0 + t + 1) * TILE_ELEMS;
                const int nb = (t + 1) & 1;
                tdm_load_tile(pred + bn, (uint32_t)(uintptr_t)&s_pred[nb][0]);
                tdm_load_tile(gt   + bn, (uint32_t)(uintptr_t)&s_gt[nb][0]);
                __builtin_amdgcn_s_wait_tensorcnt(2);
            } else {
                __builtin_amdgcn_s_wait_tensorcnt(0);
            }
            asm volatile("" ::: "memory");
        }
        __syncthreads();   // publish buffer `buf` to all waves

        acc += cell_loss(&s_pred[buf][threadIdx.x * NCH],
                         &s_gt[buf][threadIdx.x * NCH]);

        __syncthreads();   // all waves done with `buf` before it is refilled
    }

    // wave32 reduction, then cross-wave via LDS
#pragma unroll
    for (int off = 16; off > 0; off >>= 1) acc += __shfl_xor(acc, off, 32);

    const int lane = threadIdx.x & 31;
    const int wid  = threadIdx.x >> 5;
    if (lane == 0) s_part[wid] = acc;
    __syncthreads();
    if (threadIdx.x == 0) {
        float s = 0.0f;
#pragma unroll
        for (int w = 0; w < TCELLS / 32; ++w) s += s_part[w];
        ws[blockIdx.x] = s;
    }
}

__global__ void yolo_loss_finish(const float* __restrict__ ws,
                                 float* __restrict__ out, int n) {
    __shared__ float s_part[8];
    float s = 0.0f;
    for (int i = threadIdx.x; i < n; i += blockDim.x) s += ws[i];
#pragma unroll
    for (int off = 16; off > 0; off >>= 1) s += __shfl_xor(s, off, 32);
    const int lane = threadIdx.x & 31;
    const int wid  = threadIdx.x >> 5;
    if (lane == 0) s_part[wid] = s;
    __syncthreads();
    if (threadIdx.x == 0) {
        float t = 0.0f;
#pragma unroll
        for (int w = 0; w < 8; ++w) t += s_part[w];
        out[0] = t * INV_BATCH;
    }
}

extern "C" void kernel_launch(void* const* d_in, const int* in_sizes, int n_in,
                              void* d_out, int out_size, void* d_ws, size_t ws_size,
                              hipStream_t stream) {
    const float* pred = (const float*)d_in[0];
    const float* gt   = (const float*)d_in[1];
    float* ws = (float*)d_ws;
    yolo_loss_partial<<<NBLOCKS, TCELLS, 0, stream>>>(pred, gt, ws);
    yolo_loss_finish<<<1, 256, 0, stream>>>(ws, (float*)d_out, NBLOCKS);
}